// NodeFormerConv_64235530879068
// MI455X (gfx1250) — compile-verified
//
#include <hip/hip_runtime.h>
#include <hip/hip_bf16.h>
#include <math.h>

typedef float v2f __attribute__((ext_vector_type(2)));
typedef float v8f __attribute__((ext_vector_type(8)));

// D = A(16x4) * B(4x16) + C, exact fp32 WMMA on gfx1250.
#define WMMA_F32(a, b, c) \
  __builtin_amdgcn_wmma_f32_16x16x4_f32(false, (a), false, (b), (short)0, (c), false, false)

// ---- gfx1250 async global->LDS copies (ASYNCcnt path), inline asm for
// toolchain portability. LDS address = low 32 bits of generic pointer.
__device__ __forceinline__ void async_g2l_b128(const float* gsrc, float* ldst) {
  unsigned lo = (unsigned)(uintptr_t)ldst;
  asm volatile("global_load_async_to_lds_b128 %0, %1, off"
               :: "v"(lo), "v"((const void*)gsrc) : "memory");
}
__device__ __forceinline__ void async_g2l_b32(const float* gsrc, float* ldst) {
  unsigned lo = (unsigned)(uintptr_t)ldst;
  asm volatile("global_load_async_to_lds_b32 %0, %1, off"
               :: "v"(lo), "v"((const void*)gsrc) : "memory");
}
__device__ __forceinline__ void async_wait0() {
  asm volatile("s_wait_asynccnt 0x0" ::: "memory");
}

namespace nf {
constexpr int N = 30000, DIN = 128, H = 4, D = 64, M = 64, K = 6, E = 480000;
constexpr int HD = H * D;   // 256
constexpr int HM = H * M;   // 256
constexpr float EPS = 1e-6f;
constexpr float INV_TAU = 4.0f;                 // 1/0.25
constexpr float QK_SCALE = 0.7071067811865476f; // inv_sqrt_tau * D^-0.25
constexpr float RATIO = 0.125f;                 // M^-0.5
constexpr int KVS_SPLIT = 16;
constexpr int KPS_BLOCKS = 64;
} // namespace nf

__device__ __forceinline__ unsigned fkey(float f) {
  int b = __float_as_int(f);
  return (b >= 0) ? ((unsigned)b | 0x80000000u) : (~(unsigned)b);
}
__device__ __forceinline__ float fdecode(unsigned k) {
  int b = (k & 0x80000000u) ? (int)(k & 0x7fffffffu) : (int)(~k);
  return __int_as_float(b);
}

// ---------------- K0: init per-head max keys ----------------
__global__ void init_kmax_kernel(unsigned* kmax) {
  if (threadIdx.x < nf::H) kmax[threadIdx.x] = 0u; // lowest key == -inf
}

// ---------------- K1: fused QKV projection GEMM ----------------
__global__ __launch_bounds__(256) void qkv_kernel(
    const float* __restrict__ z, const float* __restrict__ Wq,
    const float* __restrict__ bq, const float* __restrict__ Wk,
    const float* __restrict__ bk, const float* __restrict__ Wv,
    const float* __restrict__ bv, float* __restrict__ qs,
    float* __restrict__ ks, float* __restrict__ vv) {
  using namespace nf;
  constexpr int LDA = DIN + 4; // 132 (x4B = 528 = 16B multiple)
  __shared__ alignas(16) float As[32 * LDA];
  __shared__ alignas(16) float Bs[64 * LDA];
  const int tid = threadIdx.x;
  const int n0 = blockIdx.x * 32;
  const int c0 = blockIdx.y * 64;     // 0..767
  const int mat = c0 >> 8;            // 0:q 1:k 2:v
  const int cm0 = c0 & 255;
  const float* W = (mat == 0) ? Wq : (mat == 1) ? Wk : Wv;
  const float* bias = (mat == 0) ? bq : (mat == 1) ? bk : bv;
  float* out = (mat == 0) ? qs : (mat == 1) ? ks : vv;
  const float scale = (mat == 2) ? 1.0f : QK_SCALE;

  // A: 32x128 rows of z, async b128 chunks (4 per thread)
  for (int i = 0; i < 4; ++i) {
    int idx = tid + i * 256;                 // chunk id, 0..1023
    int r = idx >> 5, c = (idx & 31) << 2;   // 32 chunks per row
    int gn = n0 + r;
    float* dst = &As[r * LDA + c];
    if (gn < N) {
      async_g2l_b128(&z[(size_t)gn * DIN + c], dst);
    } else {
      dst[0] = 0.f; dst[1] = 0.f; dst[2] = 0.f; dst[3] = 0.f;
    }
  }
  // B^T[n][kk] = W[kk][cm0+n]: transposed scatter via async b32
  for (int i = 0; i < 32; ++i) {
    int idx = tid + i * 256, kk = idx >> 6, nn = idx & 63;
    async_g2l_b32(&W[kk * HD + cm0 + nn], &Bs[nn * LDA + kk]);
  }
  async_wait0();
  __syncthreads();

  const int wave = tid >> 5, lane = tid & 31;
  const int half = lane >> 4, l16 = lane & 15;
  const int wr = wave >> 2, wc = wave & 3;
  const float* arow = &As[(wr * 16 + l16) * LDA];
  const float* brow = &Bs[(wc * 16 + l16) * LDA];
  v8f acc = {0.f, 0.f, 0.f, 0.f, 0.f, 0.f, 0.f, 0.f};
  for (int kk = 0; kk < DIN; kk += 4) {
    v2f a = *(const v2f*)(arow + kk + half * 2);
    v2f b = *(const v2f*)(brow + kk + half * 2);
    acc = WMMA_F32(a, b, acc);
  }
  const int col = cm0 + wc * 16 + l16;
  const float bval = bias[col];
  for (int r = 0; r < 8; ++r) {
    int gn = n0 + wr * 16 + half * 8 + r;
    if (gn < N) out[(size_t)gn * HD + col] = (acc[r] + bval) * scale;
  }
}

// ---------------- K2q: q features (one pass, row max local) ----------------
// In-place legal: each (n-tile,h) block reads exactly the region it writes.
__global__ __launch_bounds__(256) void phi_q_kernel(const float* qsrc,
                                                    const float* __restrict__ proj,
                                                    float* qp) {
  using namespace nf;
  constexpr int LDA = D + 4; // 68 (x4B = 272 = 16B multiple)
  __shared__ alignas(16) float As[32 * LDA];
  __shared__ alignas(16) float Ps[64 * LDA];
  __shared__ alignas(16) float dds[32 * LDA];
  __shared__ float dmx[32];
  const int tid = threadIdx.x;
  const int n0 = blockIdx.x * 32;
  const int h = blockIdx.y;
  for (int i = 0; i < 2; ++i) {              // 512 b128 chunks
    int idx = tid + i * 256;
    int r = idx >> 4, c = (idx & 15) << 2;
    int gn = n0 + r;
    float* dst = &As[r * LDA + c];
    if (gn < N) {
      async_g2l_b128(&qsrc[(size_t)gn * HD + h * D + c], dst);
    } else {
      dst[0] = 0.f; dst[1] = 0.f; dst[2] = 0.f; dst[3] = 0.f;
    }
  }
  for (int i = 0; i < 4; ++i) {              // proj: 1024 b128 chunks
    int idx = tid + i * 256;
    int r = idx >> 4, c = (idx & 15) << 2;
    async_g2l_b128(&proj[idx << 2], &Ps[r * LDA + c]);
  }
  async_wait0();
  __syncthreads();
  const int wave = tid >> 5, lane = tid & 31;
  const int half = lane >> 4, l16 = lane & 15;
  const int wr = wave >> 2, wc = wave & 3;
  const float* arow = &As[(wr * 16 + l16) * LDA];
  const float* brow = &Ps[(wc * 16 + l16) * LDA];
  v8f acc = {0.f, 0.f, 0.f, 0.f, 0.f, 0.f, 0.f, 0.f};
  for (int kk = 0; kk < D; kk += 4) {
    v2f a = *(const v2f*)(arow + kk + half * 2);
    v2f b = *(const v2f*)(brow + kk + half * 2);
    acc = WMMA_F32(a, b, acc);
  }
  for (int r = 0; r < 8; ++r)
    dds[(wr * 16 + half * 8 + r) * LDA + wc * 16 + l16] = acc[r];
  __syncthreads();
  if (tid < 32) {
    const float* dr = &dds[tid * LDA];
    float mx = dr[0];
    for (int m = 1; m < M; ++m) mx = fmaxf(mx, dr[m]);
    const float* ar = &As[tid * LDA];
    float dg = 0.f;
    for (int c = 0; c < D; ++c) dg += ar[c] * ar[c];
    dmx[tid] = mx + 0.5f * dg; // diag + row max
  }
  __syncthreads();
  for (int i = 0; i < 8; ++i) {
    int idx = tid + i * 256, r = idx >> 6, c = idx & 63;
    int gn = n0 + r;
    if (gn < N)
      qp[(size_t)gn * HM + h * M + c] = RATIO * (expf(dds[r * LDA + c] - dmx[r]) + EPS);
  }
}

// ---------------- K2k-a: k features, raw dd + diag + global max ----------------
__global__ __launch_bounds__(256) void phi_k_dd_kernel(
    const float* ksrc, const float* __restrict__ proj, float* ddk,
    float* __restrict__ diagk, unsigned* __restrict__ kmax) {
  using namespace nf;
  constexpr int LDA = D + 4;
  __shared__ alignas(16) float As[32 * LDA];
  __shared__ alignas(16) float Ps[64 * LDA];
  __shared__ float red[256];
  const int tid = threadIdx.x;
  const int n0 = blockIdx.x * 32;
  const int h = blockIdx.y;
  for (int i = 0; i < 2; ++i) {
    int idx = tid + i * 256;
    int r = idx >> 4, c = (idx & 15) << 2;
    int gn = n0 + r;
    float* dst = &As[r * LDA + c];
    if (gn < N) {
      async_g2l_b128(&ksrc[(size_t)gn * HD + h * D + c], dst);
    } else {
      dst[0] = 0.f; dst[1] = 0.f; dst[2] = 0.f; dst[3] = 0.f;
    }
  }
  for (int i = 0; i < 4; ++i) {
    int idx = tid + i * 256;
    int r = idx >> 4, c = (idx & 15) << 2;
    async_g2l_b128(&proj[idx << 2], &Ps[r * LDA + c]);
  }
  async_wait0();
  __syncthreads();
  const int wave = tid >> 5, lane = tid & 31;
  const int half = lane >> 4, l16 = lane & 15;
  const int wr = wave >> 2, wc = wave & 3;
  const float* arow = &As[(wr * 16 + l16) * LDA];
  const float* brow = &Ps[(wc * 16 + l16) * LDA];
  v8f acc = {0.f, 0.f, 0.f, 0.f, 0.f, 0.f, 0.f, 0.f};
  for (int kk = 0; kk < D; kk += 4) {
    v2f a = *(const v2f*)(arow + kk + half * 2);
    v2f b = *(const v2f*)(brow + kk + half * 2);
    acc = WMMA_F32(a, b, acc);
  }
  float mx8 = -3.4e38f;
  for (int r = 0; r < 8; ++r) {
    int gn = n0 + wr * 16 + half * 8 + r;
    if (gn < N) {
      ddk[(size_t)gn * HM + h * M + wc * 16 + l16] = acc[r];
      mx8 = fmaxf(mx8, acc[r]);
    }
  }
  red[tid] = mx8;
  __syncthreads();
  for (int s = 128; s > 0; s >>= 1) {
    if (tid < s) red[tid] = fmaxf(red[tid], red[tid + s]);
    __syncthreads();
  }
  if (tid == 0) atomicMax(&kmax[h], fkey(red[0]));
  if (tid < 32) {
    int gn = n0 + tid;
    if (gn < N) {
      const float* ar = &As[tid * LDA];
      float dg = 0.f;
      for (int c = 0; c < D; ++c) dg += ar[c] * ar[c];
      diagk[(size_t)gn * H + h] = 0.5f * dg;
    }
  }
}

// ---------------- K2k-b: exp transform (in place) ----------------
__global__ __launch_bounds__(256) void phi_k_exp_kernel(
    float* kp, const float* __restrict__ diagk, const unsigned* __restrict__ kmax) {
  using namespace nf;
  const int t = threadIdx.x; // column 0..255
  const int n0 = blockIdx.x * 32;
  const int h = t >> 6;
  const float mx = fdecode(kmax[h]);
  for (int i = 0; i < 32; ++i) {
    int gn = n0 + i;
    if (gn >= N) break;
    float dd = kp[(size_t)gn * HM + t];
    kp[(size_t)gn * HM + t] = RATIO * (expf(dd - diagk[(size_t)gn * H + h] - mx) + EPS);
  }
}

// ---------------- kp_sum partials (deterministic) ----------------
__global__ __launch_bounds__(256) void kpsum_part_kernel(const float* __restrict__ kp,
                                                         float* __restrict__ part) {
  using namespace nf;
  const int t = threadIdx.x, b = blockIdx.x;
  const int rows = (N + KPS_BLOCKS - 1) / KPS_BLOCKS;
  float s = 0.f;
  for (int i = 0; i < rows; ++i) {
    int n = b * rows + i;
    if (n >= N) break;
    s += kp[(size_t)n * HM + t];
  }
  part[b * HM + t] = s;
}

// ---------------- K3: kvs = sum_n (kp*eg)^T v, split-K WMMA ----------------
// (A tile needs the kp*exp(g/tau) multiply during fill -> VGPR staging kept.)
__global__ __launch_bounds__(256) void kvs_kernel(
    const float* __restrict__ kp, const float* __restrict__ vv,
    const float* __restrict__ gum, float* __restrict__ kvs_part,
    float* __restrict__ kssum_part) {
  using namespace nf;
  constexpr int LDT = M + 4; // 68
  __shared__ alignas(16) float As[64 * LDT]; // [n][m] = kp*eg
  __shared__ alignas(16) float Bs[64 * LDT]; // [n][d] = v
  __shared__ float egs[64];
  const int tid = threadIdx.x;
  const int split = blockIdx.x, h = blockIdx.y, kg = blockIdx.z;
  const int wave = tid >> 5, lane = tid & 31;
  const int half = lane >> 4, l16 = lane & 15;
  const int wrA = wave >> 2, wc = wave & 3;
  const int m0a = wrA * 16, m0b = wrA * 16 + 32;
  v8f acc0 = {0.f, 0.f, 0.f, 0.f, 0.f, 0.f, 0.f, 0.f};
  v8f acc1 = {0.f, 0.f, 0.f, 0.f, 0.f, 0.f, 0.f, 0.f};
  float kss = 0.f;
  const int nchunks = (N + 63) / 64;
  for (int c = split; c < nchunks; c += KVS_SPLIT) {
    const int cn0 = c * 64;
    __syncthreads(); // protect previous tiles
    if (tid < 64) {
      int gn = cn0 + tid;
      egs[tid] = (gn < N) ? expf(gum[(size_t)gn * (H * K) + h * K + kg] * INV_TAU) : 0.f;
    }
    __syncthreads();
    for (int i = 0; i < 16; ++i) {
      int idx = tid + i * 256, r = idx >> 6, cc = idx & 63;
      int gn = cn0 + r;
      float kv = 0.f, vval = 0.f;
      if (gn < N) {
        kv = kp[(size_t)gn * HM + h * M + cc];
        vval = vv[(size_t)gn * HD + h * D + cc];
      }
      As[r * LDT + cc] = kv * egs[r];
      Bs[r * LDT + cc] = vval;
    }
    __syncthreads();
    for (int kk = 0; kk < 64; kk += 4) { // contraction index = n
      int kb = kk + half * 2;
      v2f a0, a1, b;
      a0.x = As[kb * LDT + m0a + l16];
      a0.y = As[(kb + 1) * LDT + m0a + l16];
      a1.x = As[kb * LDT + m0b + l16];
      a1.y = As[(kb + 1) * LDT + m0b + l16];
      b.x = Bs[kb * LDT + wc * 16 + l16];
      b.y = Bs[(kb + 1) * LDT + wc * 16 + l16];
      acc0 = WMMA_F32(a0, b, acc0);
      acc1 = WMMA_F32(a1, b, acc1);
    }
    if (tid < 64) {
      for (int r = 0; r < 64; ++r) kss += As[r * LDT + tid]; // ks_sum column
    }
  }
  float* base = &kvs_part[((size_t)split * (H * K) + h * K + kg) * (M * D)];
  const int dcol = wc * 16 + l16;
  for (int r = 0; r < 8; ++r) {
    base[(m0a + half * 8 + r) * D + dcol] = acc0[r];
    base[(m0b + half * 8 + r) * D + dcol] = acc1[r];
  }
  if (tid < 64)
    kssum_part[((size_t)split * (H * K) + h * K + kg) * M + tid] = kss;
}

// ---------------- generic partial reduce: dst[i] = sum_p src[p*count+i] ----------------
__global__ void reduce_parts_kernel(const float* __restrict__ src,
                                    float* __restrict__ dst, int count, int parts) {
  int i = blockIdx.x * 256 + threadIdx.x;
  if (i >= count) return;
  float s = 0.f;
  for (int p = 0; p < parts; ++p) s += src[(size_t)p * count + i];
  dst[i] = s;
}

// ---------------- K4: z_out = mean_k (qp@kvs[h,k]) / (qp.ks_sum[h,k]) ----------------
__global__ __launch_bounds__(256) void zout_kernel(
    const float* __restrict__ qp, const float* __restrict__ kvs,
    const float* __restrict__ kssum, float* __restrict__ zout) {
  using namespace nf;
  constexpr int LDT = M + 4;
  __shared__ alignas(16) float Qs[32 * LDT];
  __shared__ alignas(16) float Ks[64 * LDT];
  __shared__ float denS[32];
  const int tid = threadIdx.x;
  const int n0 = blockIdx.x * 32;
  const int h = blockIdx.y;
  for (int i = 0; i < 2; ++i) {              // qp tile, async b128
    int idx = tid + i * 256;
    int r = idx >> 4, c = (idx & 15) << 2;
    int gn = n0 + r;
    float* dst = &Qs[r * LDT + c];
    if (gn < N) {
      async_g2l_b128(&qp[(size_t)gn * HM + h * M + c], dst);
    } else {
      dst[0] = 0.f; dst[1] = 0.f; dst[2] = 0.f; dst[3] = 0.f;
    }
  }
  const int wave = tid >> 5, lane = tid & 31;
  const int half = lane >> 4, l16 = lane & 15;
  const int wr = wave >> 2, wc = wave & 3;
  const float* arow = &Qs[(wr * 16 + l16) * LDT];
  float zacc[8] = {0.f, 0.f, 0.f, 0.f, 0.f, 0.f, 0.f, 0.f};
  for (int kg = 0; kg < K; ++kg) {
    const float* kv = &kvs[((size_t)h * K + kg) * (M * D)];
    for (int i = 0; i < 4; ++i) {            // kvs tile, async b128
      int idx = tid + i * 256;
      int r = idx >> 4, c = (idx & 15) << 2;
      async_g2l_b128(&kv[idx << 2], &Ks[r * LDT + c]);
    }
    async_wait0();
    __syncthreads(); // Qs/Ks ready; also fences previous-iter Ks/denS reuse
    if (tid < 32) {
      const float* kss = &kssum[((size_t)h * K + kg) * M];
      const float* qr = &Qs[tid * LDT];
      float dsum = 0.f;
      for (int m = 0; m < M; ++m) dsum += qr[m] * kss[m];
      denS[tid] = dsum;
    }
    __syncthreads();
    v8f acc = {0.f, 0.f, 0.f, 0.f, 0.f, 0.f, 0.f, 0.f};
    for (int kk = 0; kk < M; kk += 4) {
      int kb = kk + half * 2;
      v2f a = *(const v2f*)(arow + kb);
      v2f b;
      b.x = Ks[kb * LDT + wc * 16 + l16];
      b.y = Ks[(kb + 1) * LDT + wc * 16 + l16];
      acc = WMMA_F32(a, b, acc);
    }
    for (int r = 0; r < 8; ++r) zacc[r] += acc[r] / denS[wr * 16 + half * 8 + r];
    __syncthreads(); // done reading Ks/denS before next-iter overwrite
  }
  const float invk = 1.0f / K;
  for (int r = 0; r < 8; ++r) {
    int gn = n0 + wr * 16 + half * 8 + r;
    if (gn < N) zout[(size_t)gn * HD + h * D + wc * 16 + l16] = zacc[r] * invk;
  }
}

// ---------------- K5: z_next = z_out @ Wo + bo ----------------
__global__ __launch_bounds__(256) void proj_out_kernel(
    const float* __restrict__ zout, const float* __restrict__ Wo,
    const float* __restrict__ bo, float* __restrict__ out) {
  using namespace nf;
  constexpr int LDT = 68;
  __shared__ alignas(16) float As[32 * LDT];
  __shared__ alignas(16) float Bs[64 * LDT];
  const int tid = threadIdx.x;
  const int n0 = blockIdx.x * 32;
  const int wave = tid >> 5, lane = tid & 31;
  const int half = lane >> 4, l16 = lane & 15;
  const int wr = wave >> 2, wc = wave & 3;
  v8f acc = {0.f, 0.f, 0.f, 0.f, 0.f, 0.f, 0.f, 0.f};
  for (int kc = 0; kc < 4; ++kc) { // K=256 in 64-chunks
    __syncthreads();
    for (int i = 0; i < 2; ++i) {            // A chunk, async b128
      int idx = tid + i * 256;
      int r = idx >> 4, c = (idx & 15) << 2;
      int gn = n0 + r;
      float* dst = &As[r * LDT + c];
      if (gn < N) {
        async_g2l_b128(&zout[(size_t)gn * HD + kc * 64 + c], dst);
      } else {
        dst[0] = 0.f; dst[1] = 0.f; dst[2] = 0.f; dst[3] = 0.f;
      }
    }
    for (int i = 0; i < 16; ++i) { // B^T[d][kk] = Wo[kc*64+kk][d], async b32 scatter
      int idx = tid + i * 256, kk = idx >> 6, dd = idx & 63;
      async_g2l_b32(&Wo[(kc * 64 + kk) * D + dd], &Bs[dd * LDT + kk]);
    }
    async_wait0();
    __syncthreads();
    const float* arow = &As[(wr * 16 + l16) * LDT];
    const float* brow = &Bs[(wc * 16 + l16) * LDT];
    for (int kk = 0; kk < 64; kk += 4) {
      v2f a = *(const v2f*)(arow + kk + half * 2);
      v2f b = *(const v2f*)(brow + kk + half * 2);
      acc = WMMA_F32(a, b, acc);
    }
  }
  const int dcol = wc * 16 + l16;
  const float bv = bo[dcol];
  for (int r = 0; r < 8; ++r) {
    int gn = n0 + wr * 16 + half * 8 + r;
    if (gn < N) out[(size_t)gn * D + dcol] = acc[r] + bv;
  }
}

// ---------------- K6: norm[n,h] = qp[n,h,:] . kp_sum[h,:] ----------------
__global__ __launch_bounds__(256) void norm_kernel(const float* __restrict__ qp,
                                                   const float* __restrict__ kpsum,
                                                   float* __restrict__ nrm) {
  using namespace nf;
  const int wave = threadIdx.x >> 5, lane = threadIdx.x & 31;
  const int p = blockIdx.x * 8 + wave;
  if (p >= N * H) return;
  const int n = p >> 2, h = p & 3;
  const float* q = &qp[(size_t)n * HM + h * M];
  const float* s = &kpsum[h * M];
  float v = q[lane] * s[lane] + q[lane + 32] * s[lane + 32];
  for (int m = 16; m >= 1; m >>= 1) v += __shfl_xor(v, m, 32);
  if (lane == 0) nrm[(size_t)n * H + h] = v;
}

// ---------------- K7: edge attention weights (wave per edge) ----------------
__global__ __launch_bounds__(256) void edge_kernel(
    const int* __restrict__ ei, const float* __restrict__ qp,
    const float* __restrict__ kp, const float* __restrict__ nrm,
    float* __restrict__ aw) {
  using namespace nf;
  const int wave = threadIdx.x >> 5, lane = threadIdx.x & 31;
  const int e = blockIdx.x * 8 + wave;
  if (e >= E) return;
  const int s = ei[e], t = ei[E + e]; // start, end
  const float* q = &qp[(size_t)t * HM];
  const float* kk = &kp[(size_t)s * HM];
  for (int h = 0; h < H; ++h) {
    float v = q[h * M + lane] * kk[h * M + lane] +
              q[h * M + 32 + lane] * kk[h * M + 32 + lane];
    for (int m = 16; m >= 1; m >>= 1) v += __shfl_xor(v, m, 32);
    if (lane == 0) aw[(size_t)e * H + h] = v / nrm[(size_t)t * H + h];
  }
}

extern "C" void kernel_launch(void* const* d_in, const int* in_sizes, int n_in,
                              void* d_out, int out_size, void* d_ws, size_t ws_size,
                              hipStream_t stream) {
  using namespace nf;
  (void)in_sizes; (void)n_in; (void)out_size; (void)ws_size;
  const float* z    = (const float*)d_in[0];
  const float* Wq_w = (const float*)d_in[1];
  const float* Wq_b = (const float*)d_in[2];
  const float* Wk_w = (const float*)d_in[3];
  const float* Wk_b = (const float*)d_in[4];
  const float* Wv_w = (const float*)d_in[5];
  const float* Wv_b = (const float*)d_in[6];
  const float* Wo_w = (const float*)d_in[7];
  const float* Wo_b = (const float*)d_in[8];
  const float* proj = (const float*)d_in[9];
  const float* gum  = (const float*)d_in[10];
  const int*   ei   = (const int*)d_in[11];

  float* ws = (float*)d_ws;
  size_t off = 0;
  float* bufQ = ws + off; off += (size_t)N * HD;       // qs -> qp (in place)
  float* bufK = ws + off; off += (size_t)N * HD;       // ks -> dd -> kp (in place)
  float* bufV = ws + off; off += (size_t)N * HD;       // v  -> z_out (reuse)
  float* diagk = ws + off; off += (size_t)N * H;
  float* nrm   = ws + off; off += (size_t)N * H;
  float* kvs   = ws + off; off += (size_t)H * K * M * D;
  float* kssum = ws + off; off += (size_t)H * K * M;
  float* kpsum = ws + off; off += HM;
  unsigned* kmax = (unsigned*)(ws + off); off += 8;
  float* kvs_part = ws + off; off += (size_t)KVS_SPLIT * H * K * M * D;
  float* kss_part = ws + off; off += (size_t)KVS_SPLIT * H * K * M;
  float* kps_part = ws + off; off += (size_t)KPS_BLOCKS * HM;
  // total ~= 24.9M floats (~100 MB) of d_ws

  const int NB32 = (N + 31) / 32;

  init_kmax_kernel<<<1, 32, 0, stream>>>(kmax);
  qkv_kernel<<<dim3(NB32, 12), 256, 0, stream>>>(z, Wq_w, Wq_b, Wk_w, Wk_b,
                                                 Wv_w, Wv_b, bufQ, bufK, bufV);
  phi_q_kernel<<<dim3(NB32, H), 256, 0, stream>>>(bufQ, proj, bufQ);
  phi_k_dd_kernel<<<dim3(NB32, H), 256, 0, stream>>>(bufK, proj, bufK, diagk, kmax);
  phi_k_exp_kernel<<<dim3(NB32), 256, 0, stream>>>(bufK, diagk, kmax);
  kpsum_part_kernel<<<dim3(KPS_BLOCKS), 256, 0, stream>>>(bufK, kps_part);
  reduce_parts_kernel<<<dim3(1), 256, 0, stream>>>(kps_part, kpsum, HM, KPS_BLOCKS);
  kvs_kernel<<<dim3(KVS_SPLIT, H, K), 256, 0, stream>>>(bufK, bufV, gum,
                                                        kvs_part, kss_part);
  reduce_parts_kernel<<<dim3((H * K * M * D + 255) / 256), 256, 0, stream>>>(
      kvs_part, kvs, H * K * M * D, KVS_SPLIT);
  reduce_parts_kernel<<<dim3((H * K * M + 255) / 256), 256, 0, stream>>>(
      kss_part, kssum, H * K * M, KVS_SPLIT);
  zout_kernel<<<dim3(NB32, H), 256, 0, stream>>>(bufQ, kvs, kssum, bufV);
  proj_out_kernel<<<dim3(NB32), 256, 0, stream>>>(bufV, Wo_w, Wo_b, (float*)d_out);
  norm_kernel<<<dim3((N * H + 7) / 8), 256, 0, stream>>>(bufQ, kpsum, nrm);
  edge_kernel<<<dim3((E + 7) / 8), 256, 0, stream>>>(
      ei, bufQ, bufK, nrm, (float*)d_out + (size_t)N * D);
}